// UpdateZ_85761906966758
// MI455X (gfx1250) — compile-verified
//
#include <hip/hip_runtime.h>
#include <math.h>

// ---------------------------------------------------------------------------
// CDP phase-retrieval update for MI455X (gfx1250, wave32).
// fft2 / ifft2 are done as batched complex GEMMs with the 256x256 DFT matrix
// using V_WMMA_F32_16X16X4_F32 (the only fp32 WMMA shape on CDNA5).
// Each wave register-blocks a 32x32 output tile (2x2 WMMA tiles) so every
// fragment load feeds 4 WMMAs; conjugation signs and the fused Ytr multiply
// are template parameters so the K-loop has no branches or sign multiplies.
// ---------------------------------------------------------------------------

#define NN       256
#define NPIX     (NN * NN)        // 65536
#define BB       8
#define LL       16

typedef __attribute__((ext_vector_type(2))) float v2f;
typedef __attribute__((ext_vector_type(8))) float v8f;

// ---- fp32 WMMA wrapper ----------------------------------------------------
__device__ __forceinline__ v8f wmma_f32x4(v2f a, v2f b, v8f c) {
#if __has_builtin(__builtin_amdgcn_wmma_f32_16x16x4_f32)
    // 8 args: (neg_a, A, neg_b, B, c_mod, C, reuse_a, reuse_b)
    return __builtin_amdgcn_wmma_f32_16x16x4_f32(
        false, a, false, b, (short)0, c, false, false);
#else
#warning "wmma_f32_16x16x4_f32 builtin unavailable - emitting stub (no wmma)"
    c[0] += a.x * b.x;  // placeholder so the file still compiles
    return c;
#endif
}

// ---- DFT matrix init: F[u*N+x] = exp(-2*pi*i*u*x/N) -----------------------
__global__ void initF_kernel(float* __restrict__ Fre, float* __restrict__ Fim) {
    int idx = blockIdx.x * blockDim.x + threadIdx.x;   // 0..65535
    int u = idx >> 8, x = idx & 255;
    int t = (u * x) & 255;                             // exact mod-256 phase
    float ang = -6.283185307179586f * (float)t * (1.0f / 256.0f);
    Fre[idx] = cosf(ang);
    Fim[idx] = sinf(ang);
}

// ---- MZ = masks * Z (complex), per image chunk ----------------------------
__global__ void maskmul_kernel(const float* __restrict__ Zre, const float* __restrict__ Zim,
                               const float* __restrict__ Mre, const float* __restrict__ Mim,
                               float* __restrict__ Ore, float* __restrict__ Oim,
                               int b0) {
    int p = blockIdx.x * blockDim.x + threadIdx.x;     // pixel
    int i = blockIdx.y;                                // local image
    int gimg = b0 * LL + i;
    int b = gimg >> 4, l = gimg & 15;
    float zr = Zre[(size_t)b * NPIX + p], zi = Zim[(size_t)b * NPIX + p];
    float mr = Mre[(size_t)l * NPIX + p], mi = Mim[(size_t)l * NPIX + p];
    Ore[(size_t)i * NPIX + p] = mr * zr - mi * zi;
    Oim[(size_t)i * NPIX + p] = mr * zi + mi * zr;
}

// ---- Batched 256x256x256 complex GEMM -------------------------------------
// C = (A_re + AS*i*A_im) x (B'), B' = [Ymul .*] (B_re + BS*i*B_im), *scale.
// One wave owns a 32x32 C block = 2x2 WMMA tiles; K-loop 64 steps, 16 WMMA
// per step.  AS/BS in {+1,-1} implement conj(F) at zero runtime cost.
template <int AS, int BS, bool HASY>
__global__ __launch_bounds__(128)
void cgemm256c_kernel(const float* __restrict__ Are, const float* __restrict__ Aim,
                      long long aStride,
                      const float* __restrict__ Bre, const float* __restrict__ Bim,
                      long long bStride,
                      const float* __restrict__ Ymul, long long yStride,
                      float* __restrict__ Cre, float* __restrict__ Cim,
                      float scale) {
    const int wave = threadIdx.x >> 5;
    const int lane = threadIdx.x & 31;
    const int half = lane >> 4;          // selects K pair (ISA f32 A/B layout)
    const int lid  = lane & 15;
    const int tn2  = blockIdx.x;                 // 0..7 : 32-col block
    const int tm2  = blockIdx.y * 4 + wave;      // 0..7 : 32-row block
    const int img  = blockIdx.z;

    const float* are = Are + (long long)img * aStride;
    const float* aim = Aim + (long long)img * aStride;
    const float* bre = Bre + (long long)img * bStride;
    const float* bim = Bim + (long long)img * bStride;
    const float* ym  = HASY ? (Ymul + (long long)img * yStride) : (const float*)0;

    const int arow0 = tm2 * 32 + lid;            // A rows this lane owns
    const int bcol0 = tn2 * 32 + lid;            // B cols this lane owns

    v8f cre[2][2], cim[2][2];
#pragma unroll
    for (int i = 0; i < 2; ++i)
#pragma unroll
        for (int j = 0; j < 2; ++j) {
            cre[i][j] = (v8f){0.f,0.f,0.f,0.f,0.f,0.f,0.f,0.f};
            cim[i][j] = (v8f){0.f,0.f,0.f,0.f,0.f,0.f,0.f,0.f};
        }

    for (int k0 = 0; k0 < NN; k0 += 4) {
        const int ka = k0 + 2 * half;            // this lane's K pair base
        v2f aRe[2], aIm[2], aImN[2];
#pragma unroll
        for (int i = 0; i < 2; ++i) {
            const size_t off = (size_t)(arow0 + 16 * i) * NN + ka;
            aRe[i] = *(const v2f*)(are + off);
            v2f t  = *(const v2f*)(aim + off);
            if (AS < 0) t = -t;                  // conj(A): compile-time xor
            aIm[i]  = t;
            aImN[i] = -t;                        // fp32 WMMA has no A/B NEG
        }
        v2f bRe[2], bIm[2];
#pragma unroll
        for (int j = 0; j < 2; ++j) {
            const size_t o0 = (size_t)ka * NN + (bcol0 + 16 * j);
            float br0 = bre[o0], br1 = bre[o0 + NN];
            float bi0 = bim[o0], bi1 = bim[o0 + NN];
            if (HASY) {                          // fused Ytr .* AZ
                float y0 = ym[o0], y1 = ym[o0 + NN];
                br0 *= y0; br1 *= y1; bi0 *= y0; bi1 *= y1;
            }
            if (BS < 0) { bi0 = -bi0; bi1 = -bi1; }
            bRe[j] = (v2f){br0, br1};
            bIm[j] = (v2f){bi0, bi1};
        }
#pragma unroll
        for (int i = 0; i < 2; ++i)
#pragma unroll
            for (int j = 0; j < 2; ++j) {
                cre[i][j] = wmma_f32x4(aRe[i],  bRe[j], cre[i][j]);  // +re*re
                cre[i][j] = wmma_f32x4(aImN[i], bIm[j], cre[i][j]);  // -im*im
                cim[i][j] = wmma_f32x4(aRe[i],  bIm[j], cim[i][j]);  // +re*im
                cim[i][j] = wmma_f32x4(aIm[i],  bRe[j], cim[i][j]);  // +im*re
            }
    }

    float* cr = Cre + (long long)img * NPIX;
    float* ci = Cim + (long long)img * NPIX;
#pragma unroll
    for (int i = 0; i < 2; ++i)
#pragma unroll
        for (int j = 0; j < 2; ++j)
#pragma unroll
            for (int r = 0; r < 8; ++r) {        // C layout: M = r + 8*half
                int row = tm2 * 32 + 16 * i + r + 8 * half;
                int col = bcol0 + 16 * j;
                cr[(size_t)row * NN + col] = cre[i][j][r] * scale;
                ci[(size_t)row * NN + col] = cim[i][j][r] * scale;
            }
}

// ---- Znew[b] = (1/div) * sum_l conj(mask_l) * IY[b,l] ---------------------
__global__ void reduceL_kernel(const float* __restrict__ Are, const float* __restrict__ Aim,
                               const float* __restrict__ Mre, const float* __restrict__ Mim,
                               float* __restrict__ Znre, float* __restrict__ Znim,
                               int b0, float div) {
    int p = blockIdx.x * blockDim.x + threadIdx.x;
    int bb = blockIdx.y;                        // local b
    int b = b0 + bb;
    float sre = 0.f, sim = 0.f;
#pragma unroll
    for (int l = 0; l < LL; ++l) {
        size_t li = (size_t)(bb * LL + l) * NPIX + p;
        float vr = Are[li], vi = Aim[li];
        float mr = Mre[(size_t)l * NPIX + p], mi = Mim[(size_t)l * NPIX + p];
        sre += mr * vr + mi * vi;               // conj(m) * v
        sim += mr * vi - mi * vr;
    }
    Znre[(size_t)b * NPIX + p] = sre / div;
    Znim[(size_t)b * NPIX + p] = sim / div;
}

// ---- 5x5 'same' cross-correlation (XLA conv convention), 16 planes --------
__global__ void conv5x5_kernel(const float* __restrict__ Zre, const float* __restrict__ Zim,
                               const float* __restrict__ W, const float* __restrict__ bias,
                               float* __restrict__ Cre, float* __restrict__ Cim) {
    __shared__ float sw[25];
    if (threadIdx.x < 25) sw[threadIdx.x] = W[threadIdx.x];
    __syncthreads();
    int p = blockIdx.x * blockDim.x + threadIdx.x;
    int plane = blockIdx.y;                     // 0..7 re, 8..15 im
    int b = plane & 7;
    const float* in  = (plane < 8 ? Zre : Zim) + (size_t)b * NPIX;
    float*       out = (plane < 8 ? Cre : Cim) + (size_t)b * NPIX;
    int y = p >> 8, x = p & 255;
    float acc = bias[0];
#pragma unroll
    for (int ky = 0; ky < 5; ++ky) {
        int yy = y + ky - 2;
        if (yy < 0 || yy > 255) continue;
#pragma unroll
        for (int kx = 0; kx < 5; ++kx) {
            int xx = x + kx - 2;
            if (xx < 0 || xx > 255) continue;
            acc += in[(size_t)yy * NN + xx] * sw[ky * 5 + kx];
        }
    }
    out[p] = acc;
}

// ---- per-image Frobenius norm over re^2+im^2 ------------------------------
__global__ void norm_kernel(const float* __restrict__ Cre, const float* __restrict__ Cim,
                            float* __restrict__ norms) {
    __shared__ float s[256];
    int b = blockIdx.x;
    float sum = 0.f;
    for (int i = threadIdx.x; i < NPIX; i += 256) {
        float a = Cre[(size_t)b * NPIX + i];
        float c = Cim[(size_t)b * NPIX + i];
        sum += a * a + c * c;
    }
    s[threadIdx.x] = sum;
    __syncthreads();
    for (int stride = 128; stride > 0; stride >>= 1) {
        if (threadIdx.x < stride) s[threadIdx.x] += s[threadIdx.x + stride];
        __syncthreads();
    }
    if (threadIdx.x == 0) norms[b] = sqrtf(s[0]);
}

// ---- out = stack([re/fro, im/fro]) ----------------------------------------
__global__ void finalize_kernel(const float* __restrict__ Cre, const float* __restrict__ Cim,
                                const float* __restrict__ norms, float* __restrict__ out) {
    int idx = blockIdx.x * blockDim.x + threadIdx.x;  // 0..524287
    int b = idx >> 16;
    float fro = norms[b];
    out[idx]                      = Cre[idx] / fro;
    out[(size_t)BB * NPIX + idx]  = Cim[idx] / fro;
}

// ---------------------------------------------------------------------------
extern "C" void kernel_launch(void* const* d_in, const int* in_sizes, int n_in,
                              void* d_out, int out_size, void* d_ws, size_t ws_size,
                              hipStream_t stream) {
    const float* Zre = (const float*)d_in[0];   // (8,1,256,256)
    const float* Zim = (const float*)d_in[1];
    const float* Mre = (const float*)d_in[2];   // (16,256,256)
    const float* Mim = (const float*)d_in[3];
    const float* Ytr = (const float*)d_in[4];   // (8,16,256,256)
    const float* W   = (const float*)d_in[5];   // (5,5,1,1)
    const float* bpt = (const float*)d_in[6];   // (1,)
    float* out = (float*)d_out;

    // ---- workspace layout (floats) ----
    float* ws   = (float*)d_ws;
    float* Fre  = ws;                 // 65536
    float* Fim  = Fre  + NPIX;        // 65536
    float* Znre = Fim  + NPIX;        // 8*65536
    float* Znim = Znre + BB * NPIX;
    float* Cre  = Znim + BB * NPIX;
    float* Cim  = Cre  + BB * NPIX;
    float* norms = Cim + BB * NPIX;   // 16 (padded)
    float* big  = norms + 16;         // chunked A/B complex ping-pong buffers

    // choose how many b's per chunk fit in ws (deterministic in ws_size)
    size_t fixedFloats = (size_t)(big - ws);
    int nb = 8;
    while (nb > 1) {
        size_t need = (fixedFloats + (size_t)nb * LL * NPIX * 4) * sizeof(float);
        if (need <= ws_size) break;
        nb >>= 1;
    }
    size_t chunkImgs = (size_t)nb * LL;
    float* Are_ws = big;
    float* Aim_ws = Are_ws + chunkImgs * NPIX;
    float* Bre_ws = Aim_ws + chunkImgs * NPIX;
    float* Bim_ws = Bre_ws + chunkImgs * NPIX;

    // scale constants matching the reference's float32 arithmetic
    const float Mf = 1048576.0f;              // N*N*L
    const float Rf = 174763.0f;               // ceil(M/6)
    const float div = Mf * Rf;
    const float invN2 = 1.0f / (float)NPIX;   // ifft2 normalization

    dim3 blkE(256), grdE(NPIX / 256);
    initF_kernel<<<grdE, blkE, 0, stream>>>(Fre, Fim);

    for (int b0 = 0; b0 < BB; b0 += nb) {
        int imgs = nb * LL;
        // 1) MZ = masks * Z
        maskmul_kernel<<<dim3(NPIX / 256, imgs), blkE, 0, stream>>>(
            Zre, Zim, Mre, Mim, Are_ws, Aim_ws, b0);
        dim3 gG(8, 2, imgs), bG(128);
        // 2) B = F * MZ
        cgemm256c_kernel<1, 1, false><<<gG, bG, 0, stream>>>(
            Fre, Fim, 0, Are_ws, Aim_ws, NPIX,
            nullptr, 0, Bre_ws, Bim_ws, 1.0f);
        // 3) A = B * F  -> AZ = fft2(MZ)
        cgemm256c_kernel<1, 1, false><<<gG, bG, 0, stream>>>(
            Bre_ws, Bim_ws, NPIX, Fre, Fim, 0,
            nullptr, 0, Are_ws, Aim_ws, 1.0f);
        // 4) B = conj(F) * (Ytr .* AZ)   (Ytr fused into B-operand load)
        cgemm256c_kernel<-1, 1, true><<<gG, bG, 0, stream>>>(
            Fre, Fim, 0, Are_ws, Aim_ws, NPIX,
            Ytr + (size_t)b0 * LL * NPIX, NPIX, Bre_ws, Bim_ws, 1.0f);
        // 5) A = (B * conj(F)) / N^2  -> ifft2(Y)
        cgemm256c_kernel<1, -1, false><<<gG, bG, 0, stream>>>(
            Bre_ws, Bim_ws, NPIX, Fre, Fim, 0,
            nullptr, 0, Are_ws, Aim_ws, invN2);
        // 6) Znew = sum_l conj(mask)*IY / div
        reduceL_kernel<<<dim3(NPIX / 256, nb), blkE, 0, stream>>>(
            Are_ws, Aim_ws, Mre, Mim, Znre, Znim, b0, div);
    }

    // 7) conv 5x5 on re/im (16 planes)
    conv5x5_kernel<<<dim3(NPIX / 256, 16), blkE, 0, stream>>>(
        Znre, Znim, W, bpt, Cre, Cim);
    // 8) per-image Frobenius norms
    norm_kernel<<<dim3(BB), blkE, 0, stream>>>(Cre, Cim, norms);
    // 9) normalize + stack output
    finalize_kernel<<<dim3(BB * NPIX / 256), blkE, 0, stream>>>(Cre, Cim, norms, out);
}